// Portilla_Simoncelli_25177098289622
// MI455X (gfx1250) — compile-verified
//
#include <hip/hip_runtime.h>
#include <math.h>

// ---------------------------------------------------------------------------
// Portilla–Simoncelli statistics for a 1024x1024 image on gfx1250 (MI455X).
//   * 2-D FFTs: Stockham radix-2 in LDS, one workgroup per line (wave32 blocks)
//   * all fftshift handling folded into unshifted-frequency index math
//   * Gram / cross-correlation matrices via V_WMMA_F32_16X16X4_F32
//     (4 independent accumulators, 4-way unrolled K loop)
//   * reductions via wave32 shfl_xor + f32 global atomics
// ---------------------------------------------------------------------------

#define PI_F 3.14159265358979323846f
#define ANGC 0.8944271909999159f      // 2^3*3! / sqrt(4*6!)
#define GINV (2.0f/1024.0f)           // global pyramid grid: coord = f * 2/1024

typedef float v2f __attribute__((ext_vector_type(2)));
typedef float v8f __attribute__((ext_vector_type(8)));

// ------------------------------ output layout ------------------------------
#define OFF_PIX   0      // 6
#define OFF_MM    6      // 18
#define OFF_ACE   24     // 9*9*4*4 = 1296
#define OFF_SKEW  1320   // 5
#define OFF_KURT  1325   // 5
#define OFF_ACR   1330   // 9*9*5 = 405
#define OFF_C0    1735   // 4*4*5 = 80
#define OFF_CX0   1815   // 4*4*4 = 64
#define OFF_CR0   1879   // 8*8*5 = 320
#define OFF_CRX0  2199   // 8*8*4 = 256
#define OFF_VHPR  2455   // 1
#define OUT_TOTAL 2456

// scalar slots in ws
#define SC_ZERO   0
#define SC_MU     1
#define SC_VARUB  2
#define SC_VARI   3
#define SC_LOMEAN 4
#define SC_TMP    5
#define SC_PMEAN  8      // 8..11
#define SC_MM     16     // 16..33

// ------------------------------ device helpers -----------------------------
__device__ __forceinline__ float rcfun(float x) {
    float t = fminf(fmaxf(-x, 0.0f), 1.0f);
    float c = __cosf(0.5f * PI_F * t);
    return c * c;
}

__device__ __forceinline__ void fcoords(int u, int v, int n, float invHalf,
                                        float& xx, float& yy, float& lr) {
    int fx = (u < (n >> 1)) ? u : u - n;
    int fy = (v < (n >> 1)) ? v : v - n;
    xx = fx * invHalf;
    yy = fy * invHalf;
    float rad = sqrtf(xx * xx + yy * yy);
    if (fx == 0 && fy == 0) rad = invHalf;  // DC fix (pixel-left value)
    lr = log2f(rad);
}

__device__ __forceinline__ float amaskf(float xx, float yy, int b, int cplx) {
    float ang = atan2f(yy, xx);
    float a = PI_F + ang - PI_F * (float)b * 0.25f;
    a = a - 2.0f * PI_F * floorf(a * (0.5f / PI_F));   // jnp.mod semantics
    float alfa = a - PI_F;
    float c = __cosf(alfa);
    float m = ANGC * c * c * c;
    if (cplx) m = (fabsf(alfa) < 0.5f * PI_F) ? 2.0f * m : 0.0f;
    return m;
}

__device__ __forceinline__ float wredSum(float v) {
    for (int o = 16; o > 0; o >>= 1) v += __shfl_xor(v, o, 32);
    return v;
}
__device__ __forceinline__ float wredMin(float v) {
    for (int o = 16; o > 0; o >>= 1) v = fminf(v, __shfl_xor(v, o, 32));
    return v;
}
__device__ __forceinline__ float wredMax(float v) {
    for (int o = 16; o > 0; o >>= 1) v = fmaxf(v, __shfl_xor(v, o, 32));
    return v;
}
__device__ __forceinline__ unsigned fkey(float f) {
    unsigned u = __float_as_uint(f);
    return (u & 0x80000000u) ? ~u : (u | 0x80000000u);
}
__device__ __forceinline__ float funkey(unsigned k) {
    return (k & 0x80000000u) ? __uint_as_float(k ^ 0x80000000u)
                             : __uint_as_float(~k);
}

// ------------------------------ FFT (Stockham) -----------------------------
template <int N>
__global__ __launch_bounds__(N / 2) void k_fft_line(float2* __restrict__ data,
                                                    int stride, int lineStride,
                                                    int dir, float scale) {
    __shared__ float2 sA[N];
    __shared__ float2 sB[N];
    const int t = threadIdx.x;
    float2* base = data + (size_t)blockIdx.x * (size_t)lineStride;
    sA[t]         = base[(size_t)t * stride];
    sA[t + N / 2] = base[(size_t)(t + N / 2) * stride];
    __syncthreads();
    float2* src = sA;
    float2* dst = sB;
    const float sgn = (float)dir;
    for (int Ns = 1; Ns < N; Ns <<= 1) {
        int k = t & (Ns - 1);
        int j = ((t - k) << 1) + k;
        float ang = sgn * PI_F * (float)k / (float)Ns;
        float sn, cs;
        __sincosf(ang, &sn, &cs);
        float2 a = src[t];
        float2 b = src[t + N / 2];
        float2 wb = make_float2(b.x * cs - b.y * sn, b.x * sn + b.y * cs);
        dst[j]      = make_float2(a.x + wb.x, a.y + wb.y);
        dst[j + Ns] = make_float2(a.x - wb.x, a.y - wb.y);
        __syncthreads();
        float2* tmp = src; src = dst; dst = tmp;
    }
    base[(size_t)t * stride]           = make_float2(src[t].x * scale, src[t].y * scale);
    base[(size_t)(t + N / 2) * stride] = make_float2(src[t + N / 2].x * scale,
                                                     src[t + N / 2].y * scale);
}

// ------------------------------ elementwise --------------------------------
__global__ void k_zero(float* __restrict__ p, long n) {
    for (long i = blockIdx.x * (long)blockDim.x + threadIdx.x; i < n;
         i += (long)gridDim.x * blockDim.x) p[i] = 0.0f;
}

__global__ void k_r2c_sub(const float* __restrict__ src, long stride,
                          const float* __restrict__ meanPtr,
                          float2* __restrict__ dst, long n) {
    float mu = *meanPtr;
    for (long i = blockIdx.x * (long)blockDim.x + threadIdx.x; i < n;
         i += (long)gridDim.x * blockDim.x)
        dst[i] = make_float2(src[i * stride] - mu, 0.0f);
}

__global__ void k_c2r(float* __restrict__ dst, const float2* __restrict__ src, long n) {
    for (long i = blockIdx.x * (long)blockDim.x + threadIdx.x; i < n;
         i += (long)gridDim.x * blockDim.x) dst[i] = src[i].x;
}

__global__ void k_add_real(float* __restrict__ dst, const float2* __restrict__ src, long n) {
    for (long i = blockIdx.x * (long)blockDim.x + threadIdx.x; i < n;
         i += (long)gridDim.x * blockDim.x) dst[i] += src[i].x;
}

__global__ void k_sub_mean(float* __restrict__ dst, const float* __restrict__ src,
                           const float* __restrict__ meanPtr, long n) {
    float mu = *meanPtr;
    for (long i = blockIdx.x * (long)blockDim.x + threadIdx.x; i < n;
         i += (long)gridDim.x * blockDim.x) dst[i] = src[i] - mu;
}

__global__ void k_power(float2* __restrict__ c, long n) {
    for (long i = blockIdx.x * (long)blockDim.x + threadIdx.x; i < n;
         i += (long)gridDim.x * blockDim.x) {
        float2 v = c[i];
        c[i] = make_float2(v.x * v.x + v.y * v.y, 0.0f);
    }
}

__global__ void k_store_band(const float2* __restrict__ src, float2* __restrict__ cdst,
                             float* __restrict__ mdst, long n) {
    for (long i = blockIdx.x * (long)blockDim.x + threadIdx.x; i < n;
         i += (long)gridDim.x * blockDim.x) {
        float2 v = src[i];
        cdst[i] = v;
        mdst[i] = sqrtf(v.x * v.x + v.y * v.y);
    }
}

// ------------------------------ mask kernels -------------------------------
__global__ void k_split_hi_lo(const float2* __restrict__ F, float2* __restrict__ hi,
                              float2* __restrict__ lo, int n) {
    long n2 = (long)n * n;
    for (long i = blockIdx.x * (long)blockDim.x + threadIdx.x; i < n2;
         i += (long)gridDim.x * blockDim.x) {
        int v = (int)(i / n), u = (int)(i % n);
        float xx, yy, lr;
        fcoords(u, v, n, GINV, xx, yy, lr);
        float rc = rcfun(lr);
        float2 f = F[i];
        float h = sqrtf(rc), l = sqrtf(1.0f - rc);
        hi[i] = make_float2(f.x * h, f.y * h);
        lo[i] = make_float2(f.x * l, f.y * l);
    }
}

__global__ void k_band_dft(const float2* __restrict__ lod, float2* __restrict__ out,
                           int n, float shift, int b) {
    long n2 = (long)n * n;
    for (long i = blockIdx.x * (long)blockDim.x + threadIdx.x; i < n2;
         i += (long)gridDim.x * blockDim.x) {
        int v = (int)(i / n), u = (int)(i % n);
        float xx, yy, lr;
        fcoords(u, v, n, GINV, xx, yy, lr);
        float him = sqrtf(rcfun(lr + shift));
        float w = him * amaskf(xx, yy, b, 1);
        float2 f = lod[i];
        // cfac = (-1j)^3 = +1j :  i*(x+iy) = (-y, x)
        out[i] = make_float2(-f.y * w, f.x * w);
    }
}

__global__ void k_lo_crop(const float2* __restrict__ in, float2* __restrict__ out,
                          int nOut, float shift) {
    int nIn = nOut * 2;
    long n2 = (long)nOut * nOut;
    for (long i = blockIdx.x * (long)blockDim.x + threadIdx.x; i < n2;
         i += (long)gridDim.x * blockDim.x) {
        int v = (int)(i / nOut), u = (int)(i % nOut);
        int su = u + ((u >= (nOut >> 1)) ? nOut : 0);
        int sv = v + ((v >= (nOut >> 1)) ? nOut : 0);
        float xx, yy, lr;
        fcoords(u, v, nOut, GINV, xx, yy, lr);
        float w = sqrtf(1.0f - rcfun(lr + shift));
        float2 f = in[(long)sv * nIn + su];
        out[i] = make_float2(f.x * w, f.y * w);
    }
}

__global__ void k_mul_lo0_local(float2* __restrict__ F, int n) {
    long n2 = (long)n * n;
    float inv = 2.0f / (float)n;
    for (long i = blockIdx.x * (long)blockDim.x + threadIdx.x; i < n2;
         i += (long)gridDim.x * blockDim.x) {
        int v = (int)(i / n), u = (int)(i % n);
        float xx, yy, lr;
        fcoords(u, v, n, inv, xx, yy, lr);
        float w = sqrtf(1.0f - rcfun(lr));
        float2 f = F[i];
        F[i] = make_float2(f.x * w, f.y * w);
    }
}

__global__ void k_accum_recon(float2* __restrict__ res, const float2* __restrict__ F,
                              int n, int b) {
    long n2 = (long)n * n;
    float inv = 2.0f / (float)n;
    for (long i = blockIdx.x * (long)blockDim.x + threadIdx.x; i < n2;
         i += (long)gridDim.x * blockDim.x) {
        int v = (int)(i / n), u = (int)(i % n);
        float xx, yy, lr;
        fcoords(u, v, n, inv, xx, yy, lr);
        float w = sqrtf(rcfun(lr + 1.0f)) * amaskf(xx, yy, b, 0);
        float2 f = F[i];
        // cfac = (1j)^3 = -1j :  -i*(x+iy) = (y, -x)
        float2 r = res[i];
        res[i] = make_float2(r.x + f.y * w, r.y - f.x * w);
    }
}

__global__ void k_expand_scatter(const float2* __restrict__ small, int m,
                                 float2* __restrict__ big) {
    int n = m * 2;
    long m2 = (long)m * m;
    for (long i = blockIdx.x * (long)blockDim.x + threadIdx.x; i < m2;
         i += (long)gridDim.x * blockDim.x) {
        int v = (int)(i / m), u = (int)(i % m);
        int fx = (u < (m >> 1)) ? u : u - m;
        int fy = (v < (m >> 1)) ? v : v - m;
        int bu = (fx < 0) ? fx + n : fx;
        int bv = (fy < 0) ? fy + n : fy;
        big[(long)bv * n + bu] = small[i];
    }
}

__global__ void k_parent_cols(const float2* __restrict__ src, float* __restrict__ rr,
                              float* __restrict__ ri, float* __restrict__ mg, long n) {
    for (long i = blockIdx.x * (long)blockDim.x + threadIdx.x; i < n;
         i += (long)gridDim.x * blockDim.x) {
        float2 c = src[i];
        float m = sqrtf(c.x * c.x + c.y * c.y);
        float im = (m > 0.0f) ? 1.0f / m : 0.0f;
        // phi = atan2(real, imag); mag*cos(2phi) = (i^2-r^2)/m ; mag*sin(2phi)=2ri/m
        rr[i] = (c.y * c.y - c.x * c.x) * im;
        ri[i] = 2.0f * c.x * c.y * im;
        mg[i] = m;
    }
}

__global__ void k_make_rolled(const float* __restrict__ P, int n, float* __restrict__ out) {
    long n2 = (long)n * n;
    for (long i = blockIdx.x * (long)blockDim.x + threadIdx.x; i < n2;
         i += (long)gridDim.x * blockDim.x) {
        int y = (int)(i / n), x = (int)(i % n);
        out[i]          = P[i];
        out[n2 + i]     = P[(long)y * n + ((x + n - 1) % n)];
        out[2 * n2 + i] = P[(long)y * n + ((x + 1) % n)];
        out[3 * n2 + i] = P[(long)((y + n - 1) % n) * n + x];
        out[4 * n2 + i] = P[(long)((y + 1) % n) * n + x];
    }
}

// ------------------------------ autocorr extract ---------------------------
__global__ void k_ac_extract(const float2* __restrict__ ac, int n, float invHW,
                             float* __restrict__ base, int stride,
                             float* __restrict__ vari) {
    int t = threadIdx.x;
    if (t >= 81) return;
    int dy = t / 9 - 4, dx = t % 9 - 4;
    int v = (dy + n) & (n - 1);
    int u = (dx + n) & (n - 1);
    float val = ac[(long)v * n + u].x * invHW;
    base[(long)t * stride] = val;
    if (dy == 0 && dx == 0) *vari = val;
}

// ------------------------------ reductions ---------------------------------
__global__ void k_init_acc(float* acc) {
    acc[0] = acc[1] = acc[2] = acc[3] = acc[4] = 0.0f;
    ((unsigned*)acc)[5] = 0xFFFFFFFFu;
    ((unsigned*)acc)[6] = 0u;
}

__global__ void k_reduce(const float* __restrict__ p, long stride, long n,
                         const float* __restrict__ meanPtr, float* __restrict__ acc) {
    float mu = *meanPtr;
    float s1 = 0, s2 = 0, s3 = 0, s4 = 0, sa = 0;
    float mn = 3.402823466e38f, mx = -3.402823466e38f;
    for (long i = blockIdx.x * (long)blockDim.x + threadIdx.x; i < n;
         i += (long)gridDim.x * blockDim.x) {
        float v = p[i * stride] - mu;
        float v2 = v * v;
        s1 += v; s2 += v2; s3 += v2 * v; s4 += v2 * v2; sa += fabsf(v);
        mn = fminf(mn, v); mx = fmaxf(mx, v);
    }
    s1 = wredSum(s1); s2 = wredSum(s2); s3 = wredSum(s3);
    s4 = wredSum(s4); sa = wredSum(sa);
    mn = wredMin(mn); mx = wredMax(mx);
    if ((threadIdx.x & 31) == 0) {
        atomicAdd(&acc[0], s1); atomicAdd(&acc[1], s2);
        atomicAdd(&acc[2], s3); atomicAdd(&acc[3], s4);
        atomicAdd(&acc[4], sa);
        atomicMin(&((unsigned*)acc)[5], fkey(mn));
        atomicMax(&((unsigned*)acc)[6], fkey(mx));
    }
}

__global__ void k_sscal(float* dst, const float* src, int idx, float scale) {
    *dst = src[idx] * scale;
}

__global__ void k_fin_pix(const float* a1, const float* a2, float n,
                          float* out, float* varUbDst) {
    float mu = a1[0] / n;
    float vp = a2[1] / n;
    float vu = vp * n / (n - 1.0f);
    out[0] = mu;
    out[1] = vu;
    out[2] = (a2[2] / n) / powf(vp, 1.5f);
    out[3] = (a2[3] / n) / (vp * vp);
    out[4] = funkey(((const unsigned*)a1)[5]);
    out[5] = funkey(((const unsigned*)a1)[6]);
    *varUbDst = vu;
}

__global__ void k_fin_sk(const float* a, float n, const float* variP,
                         const float* varUbP, float* sk, float* ku) {
    float vari = *variP, vu = *varUbP;
    int cond = (vari / vu) > 1e-6f;
    float v = fmaxf(vari, 1e-12f);
    *sk = cond ? (a[2] / n) / powf(v, 1.5f) : 0.0f;
    *ku = cond ? (a[3] / n) / (v * v) : 3.0f;
}

// ------------------------------ WMMA Gram ----------------------------------
struct GramSet {
    const float* p[16];
    const float* sub[16];
    long stride[16];
};

#if __has_builtin(__builtin_amdgcn_wmma_f32_16x16x4_f32)
#define HAVE_WMMA_F32 1
__device__ __forceinline__ v8f gramStep(const float* xp, float xs, long xst,
                                        const float* yp, float ys, long yst,
                                        long k0, int same, v8f acc) {
    v2f a, b;
    a[0] = xp[k0 * xst] - xs;
    a[1] = xp[(k0 + 1) * xst] - xs;
    if (same) {
        b = a;
    } else {
        b[0] = yp[k0 * yst] - ys;
        b[1] = yp[(k0 + 1) * yst] - ys;
    }
    return __builtin_amdgcn_wmma_f32_16x16x4_f32(false, a, false, b,
                                                 (short)0, acc, false, false);
}
#endif

__global__ void k_gram(GramSet X, GramSet Y, long chunks, int same,
                       float* __restrict__ out) {
    const int lane = threadIdx.x & 31;
    const int col = lane & 15;
    const int half = lane >> 4;
    const long wave = ((long)blockIdx.x * blockDim.x + threadIdx.x) >> 5;
    const long nw = ((long)gridDim.x * blockDim.x) >> 5;

    const float* xp = X.p[col];  const float xs = *X.sub[col];  const long xst = X.stride[col];
    const float* yp = Y.p[col];  const float ys = *Y.sub[col];  const long yst = Y.stride[col];

    v8f acc0 = {};
#if HAVE_WMMA_F32
    // 4 independent accumulators: breaks the WMMA->WMMA dependency chain and
    // lets gather loads overlap the matrix pipeline.
    v8f acc1 = {}, acc2 = {}, acc3 = {};
    long ch = wave;
    const long koff = half * 2;   // lanes 0-15: K=0,1 ; lanes 16-31: K=2,3
    for (; ch + 3 * nw < chunks; ch += 4 * nw) {
        __builtin_prefetch(xp + ((ch + 4 * nw) * 4 + koff) * xst, 0, 1);
        acc0 = gramStep(xp, xs, xst, yp, ys, yst, (ch)          * 4 + koff, same, acc0);
        acc1 = gramStep(xp, xs, xst, yp, ys, yst, (ch + nw)     * 4 + koff, same, acc1);
        acc2 = gramStep(xp, xs, xst, yp, ys, yst, (ch + 2 * nw) * 4 + koff, same, acc2);
        acc3 = gramStep(xp, xs, xst, yp, ys, yst, (ch + 3 * nw) * 4 + koff, same, acc3);
    }
    for (; ch < chunks; ch += nw)
        acc0 = gramStep(xp, xs, xst, yp, ys, yst, ch * 4 + koff, same, acc0);
    acc1 += acc3;
    acc0 += acc2;
    acc0 += acc1;
#else
    // scalar fallback keeping the same D layout (M = r + half*8, N = col)
    (void)same;
    const float* xr[8]; float xrs[8]; long xrst[8];
    for (int r = 0; r < 8; ++r) {
        xr[r] = X.p[r + half * 8]; xrs[r] = *X.sub[r + half * 8];
        xrst[r] = X.stride[r + half * 8];
    }
    for (long ch = wave; ch < chunks; ch += nw) {
        for (int kk = 0; kk < 4; ++kk) {
            long k = ch * 4 + kk;
            float yv = yp[k * yst] - ys;
            for (int r = 0; r < 8; ++r)
                acc0[r] += (xr[r][k * xrst[r]] - xrs[r]) * yv;
        }
    }
#endif
    for (int r = 0; r < 8; ++r)
        atomicAdd(&out[(r + half * 8) * 16 + col], acc0[r]);
}

__global__ void k_gram_out(const float* __restrict__ g, float scale,
                           float* __restrict__ base, int rows, int cols,
                           int colStride, int rowStride) {
    int i = threadIdx.x;
    if (i >= rows * cols) return;
    int r = i / cols, c = i % cols;
    base[(long)r * rowStride + (long)c * colStride] = g[r * 16 + c] * scale;
}

// ------------------------------ host helpers -------------------------------
static inline dim3 gridFor(long n) {
    long b = (n + 255) / 256;
    if (b > 4096) b = 4096;
    if (b < 1) b = 1;
    return dim3((unsigned)b);
}

static void fft2d(float2* buf, int n, int dir, hipStream_t st) {
    float sc = (dir > 0) ? 1.0f / (float)n : 1.0f;
    switch (n) {
#define FFT_CASE(NN)                                                          \
    case NN:                                                                  \
        k_fft_line<NN><<<dim3(NN), dim3(NN / 2), 0, st>>>(buf, 1, NN, dir, sc); \
        k_fft_line<NN><<<dim3(NN), dim3(NN / 2), 0, st>>>(buf, NN, 1, dir, sc); \
        break;
        FFT_CASE(64) FFT_CASE(128) FFT_CASE(256) FFT_CASE(512) FFT_CASE(1024)
#undef FFT_CASE
        default: break;
    }
}

extern "C" void kernel_launch(void* const* d_in, const int* in_sizes, int n_in,
                              void* d_out, int out_size, void* d_ws, size_t ws_size,
                              hipStream_t stream) {
    (void)in_sizes; (void)n_in; (void)out_size; (void)ws_size;
    const float* img = (const float*)d_in[0];
    float* out = (float*)d_out;

    static const int  NS[4]  = {1024, 512, 256, 128};
    static const long N2S[4] = {1048576, 262144, 65536, 16384};

    // ---- workspace bump allocator ----
    char* w = (char*)d_ws;
    auto alloc = [&](size_t bytes) -> char* {
        char* r = w;
        w += (bytes + 255) & ~(size_t)255;
        return r;
    };
    float*  scal  = (float*)alloc(1024 * 4);
    float*  acc1  = (float*)alloc(64);
    float*  acc2  = (float*)alloc(64);
    float*  gram  = (float*)alloc(256 * 4);
    float2* cbufA = (float2*)alloc((size_t)1048576 * 8);
    float2* cbufB = (float2*)alloc((size_t)1048576 * 8);
    float2* cbufC = (float2*)alloc((size_t)1048576 * 8);
    float2* lodA  = (float2*)alloc((size_t)1048576 * 8);
    float2* lodB  = (float2*)alloc((size_t)1048576 * 8);
    long coffs[4], prefix = 0;
    for (int s = 0; s < 4; ++s) { coffs[s] = prefix; prefix += 4 * N2S[s]; }
    float2* coeffAr = (float2*)alloc((size_t)prefix * 8);
    float*  magAr   = (float*)alloc((size_t)prefix * 4);
    float*  hiPlane = (float*)alloc((size_t)1048576 * 4);
    float*  loRaw   = (float*)alloc(4096 * 4);
    float*  loPlane = (float*)alloc(4096 * 4);
    float*  imA     = (float*)alloc((size_t)1048576 * 4);
    float*  imB     = (float*)alloc((size_t)1048576 * 4);
    float*  rrAr[4], *riAr[4], *mgAr[4];
    for (int b = 0; b < 4; ++b) {
        rrAr[b] = (float*)alloc((size_t)1048576 * 4);
        riAr[b] = (float*)alloc((size_t)1048576 * 4);
        mgAr[b] = (float*)alloc((size_t)1048576 * 4);
    }
    float* rolled = (float*)alloc((size_t)16384 * 5 * 4);

    auto coeff = [&](int s, int b) { return coeffAr + coffs[s] + (long)b * N2S[s]; };
    auto magP  = [&](int s, int b) { return magAr + coffs[s] + (long)b * N2S[s]; };
    auto SCP   = [&](int i) { return scal + i; };

    // ---- init ----
    k_zero<<<gridFor(OUT_TOTAL), 256, 0, stream>>>(out, OUT_TOTAL);
    k_zero<<<1, 256, 0, stream>>>(scal, 64);

    const long N2 = 1048576;

    // ---- pixel stats ----
    k_init_acc<<<1, 1, 0, stream>>>(acc1);
    k_reduce<<<gridFor(N2), 256, 0, stream>>>(img, 1, N2, SCP(SC_ZERO), acc1);
    k_sscal<<<1, 1, 0, stream>>>(SCP(SC_MU), acc1, 0, 1.0f / (float)N2);
    k_init_acc<<<1, 1, 0, stream>>>(acc2);
    k_reduce<<<gridFor(N2), 256, 0, stream>>>(img, 1, N2, SCP(SC_MU), acc2);
    k_fin_pix<<<1, 1, 0, stream>>>(acc1, acc2, (float)N2, out + OFF_PIX, SCP(SC_VARUB));

    // ---- imdft ----
    k_r2c_sub<<<gridFor(N2), 256, 0, stream>>>(img, 1, SCP(SC_ZERO), cbufA, N2);
    fft2d(cbufA, 1024, -1, stream);
    k_split_hi_lo<<<gridFor(N2), 256, 0, stream>>>(cbufA, cbufB, lodA, 1024);

    // ---- hi residual: magMean[0] + vHPR0 ----
    fft2d(cbufB, 1024, +1, stream);
    k_c2r<<<gridFor(N2), 256, 0, stream>>>(hiPlane, cbufB, N2);
    k_init_acc<<<1, 1, 0, stream>>>(acc1);
    k_reduce<<<gridFor(N2), 256, 0, stream>>>(hiPlane, 1, N2, SCP(SC_ZERO), acc1);
    k_sscal<<<1, 1, 0, stream>>>(out + OFF_MM + 0, acc1, 4, 1.0f / (float)N2);
    k_sscal<<<1, 1, 0, stream>>>(SCP(SC_MM + 0), acc1, 4, 1.0f / (float)N2);
    k_sscal<<<1, 1, 0, stream>>>(out + OFF_VHPR, acc1, 1, 1.0f / (float)N2);

    // ---- pyramid bands ----
    float2* lodCur = lodA;
    float2* lodNxt = lodB;
    for (int s = 0; s < 4; ++s) {
        int n = NS[s];
        long n2 = N2S[s];
        for (int b = 0; b < 4; ++b) {
            k_band_dft<<<gridFor(n2), 256, 0, stream>>>(lodCur, cbufB, n, (float)(s + 1), b);
            fft2d(cbufB, n, +1, stream);
            k_store_band<<<gridFor(n2), 256, 0, stream>>>(cbufB, coeff(s, b), magP(s, b), n2);
            k_init_acc<<<1, 1, 0, stream>>>(acc1);
            k_reduce<<<gridFor(n2), 256, 0, stream>>>(magP(s, b), 1, n2, SCP(SC_ZERO), acc1);
            int mi = 1 + s * 4 + b;
            k_sscal<<<1, 1, 0, stream>>>(out + OFF_MM + mi, acc1, 0, 1.0f / (float)n2);
            k_sscal<<<1, 1, 0, stream>>>(SCP(SC_MM + mi), acc1, 0, 1.0f / (float)n2);
        }
        k_lo_crop<<<gridFor(n2 / 4), 256, 0, stream>>>(lodCur, lodNxt, n / 2, (float)(s + 1));
        float2* t = lodCur; lodCur = lodNxt; lodNxt = t;
    }

    // ---- low residual ----
    fft2d(lodCur, 64, +1, stream);
    k_c2r<<<gridFor(4096), 256, 0, stream>>>(loRaw, lodCur, 4096);
    k_init_acc<<<1, 1, 0, stream>>>(acc1);
    k_reduce<<<gridFor(4096), 256, 0, stream>>>(loRaw, 1, 4096, SCP(SC_ZERO), acc1);
    k_sscal<<<1, 1, 0, stream>>>(SCP(SC_LOMEAN), acc1, 0, 1.0f / 4096.0f);
    k_sub_mean<<<gridFor(4096), 256, 0, stream>>>(loPlane, loRaw, SCP(SC_LOMEAN), 4096);
    k_init_acc<<<1, 1, 0, stream>>>(acc1);
    k_reduce<<<gridFor(4096), 256, 0, stream>>>(loPlane, 1, 4096, SCP(SC_ZERO), acc1);
    k_sscal<<<1, 1, 0, stream>>>(out + OFF_MM + 17, acc1, 4, 1.0f / 4096.0f);
    k_sscal<<<1, 1, 0, stream>>>(SCP(SC_MM + 17), acc1, 4, 1.0f / 4096.0f);

    // ---- autocorr helper ----
    auto autocorr = [&](const float* plane, long strideIn, const float* meanPtr,
                        int n, float* dstBase, int elemStride, float* variPtr) {
        long n2 = (long)n * n;
        k_r2c_sub<<<gridFor(n2), 256, 0, stream>>>(plane, strideIn, meanPtr, cbufA, n2);
        fft2d(cbufA, n, -1, stream);
        k_power<<<gridFor(n2), 256, 0, stream>>>(cbufA, n2);
        fft2d(cbufA, n, +1, stream);
        k_ac_extract<<<1, 96, 0, stream>>>(cbufA, n, 1.0f / (float)n2,
                                           dstBase, elemStride, variPtr);
    };

    // ---- recon chain start: im = lowpass0(loPlane) ----
    k_r2c_sub<<<gridFor(4096), 256, 0, stream>>>(loPlane, 1, SCP(SC_ZERO), cbufA, 4096);
    fft2d(cbufA, 64, -1, stream);
    k_mul_lo0_local<<<gridFor(4096), 256, 0, stream>>>(cbufA, 64);
    fft2d(cbufA, 64, +1, stream);
    float* imCur = imA;
    float* imNxt = imB;
    k_c2r<<<gridFor(4096), 256, 0, stream>>>(imCur, cbufA, 4096);
    autocorr(imCur, 1, SCP(SC_ZERO), 64, out + OFF_ACR + 4, 5, SCP(SC_VARI));
    k_init_acc<<<1, 1, 0, stream>>>(acc1);
    k_reduce<<<gridFor(4096), 256, 0, stream>>>(imCur, 1, 4096, SCP(SC_ZERO), acc1);
    k_fin_sk<<<1, 1, 0, stream>>>(acc1, 4096.0f, SCP(SC_VARI), SCP(SC_VARUB),
                                  out + OFF_SKEW + 4, out + OFF_KURT + 4);

    // ---- recon loop s = 3..0 ----
    for (int s = 3; s >= 0; --s) {
        int n = NS[s], m = n / 2;
        long n2 = N2S[s], m2 = n2 / 4;
        for (int b = 0; b < 4; ++b)
            autocorr(magP(s, b), 1, SCP(SC_MM + 1 + s * 4 + b), n,
                     out + OFF_ACE + s * 4 + b, 16, SCP(SC_TMP));
        // expand im (m -> n)
        k_r2c_sub<<<gridFor(m2), 256, 0, stream>>>(imCur, 1, SCP(SC_ZERO), cbufA, m2);
        fft2d(cbufA, m, -1, stream);
        k_zero<<<gridFor(2 * n2), 256, 0, stream>>>((float*)cbufC, 2 * n2);
        k_expand_scatter<<<gridFor(m2), 256, 0, stream>>>(cbufA, m, cbufC);
        fft2d(cbufC, n, +1, stream);
        k_c2r<<<gridFor(n2), 256, 0, stream>>>(imNxt, cbufC, n2);
        // recon_level0 of real band parts
        k_zero<<<gridFor(2 * n2), 256, 0, stream>>>((float*)cbufB, 2 * n2);
        for (int b = 0; b < 4; ++b) {
            k_r2c_sub<<<gridFor(n2), 256, 0, stream>>>((const float*)coeff(s, b), 2,
                                                       SCP(SC_ZERO), cbufA, n2);
            fft2d(cbufA, n, -1, stream);
            k_accum_recon<<<gridFor(n2), 256, 0, stream>>>(cbufB, cbufA, n, b);
        }
        k_mul_lo0_local<<<gridFor(n2), 256, 0, stream>>>(cbufB, n);
        fft2d(cbufB, n, +1, stream);
        k_add_real<<<gridFor(n2), 256, 0, stream>>>(imNxt, cbufB, n2);
        float* t = imCur; imCur = imNxt; imNxt = t;
        autocorr(imCur, 1, SCP(SC_ZERO), n, out + OFF_ACR + s, 5, SCP(SC_VARI));
        k_init_acc<<<1, 1, 0, stream>>>(acc1);
        k_reduce<<<gridFor(n2), 256, 0, stream>>>(imCur, 1, n2, SCP(SC_ZERO), acc1);
        k_fin_sk<<<1, 1, 0, stream>>>(acc1, (float)n2, SCP(SC_VARI), SCP(SC_VARUB),
                                      out + OFF_SKEW + s, out + OFF_KURT + s);
    }

    // ---- correlation matrices (WMMA) ----
    auto mkSet = [&]() {
        GramSet g;
        for (int i = 0; i < 16; ++i) { g.p[i] = SCP(SC_ZERO); g.sub[i] = SCP(SC_ZERO); g.stride[i] = 0; }
        return g;
    };
    auto runGram = [&](const GramSet& X, const GramSet& Y, long csz, int same, float scale,
                       float* base, int rows, int cols, int colStride, int rowStride) {
        k_zero<<<1, 256, 0, stream>>>(gram, 256);
        k_gram<<<dim3(128), dim3(256), 0, stream>>>(X, Y, csz / 4, same, gram);
        k_gram_out<<<1, 64, 0, stream>>>(gram, scale, base, rows, cols, colStride, rowStride);
    };

    for (int s = 0; s < 4; ++s) {
        int n = NS[s], m = n / 2;
        long csz = N2S[s];
        float invC = 1.0f / (float)csz;

        GramSet Xc = mkSet();   // cousins: apyr = mag - magMean
        GramSet Xrc = mkSet();  // rcousins: Re(coeff)
        for (int b = 0; b < 4; ++b) {
            Xc.p[b] = magP(s, b);  Xc.sub[b] = SCP(SC_MM + 1 + s * 4 + b); Xc.stride[b] = 1;
            Xrc.p[b] = (const float*)coeff(s, b); Xrc.stride[b] = 2;
        }
        runGram(Xc, Xc, csz, 1, invC, out + OFF_C0 + s, 4, 4, 5, 20);
        runGram(Xrc, Xrc, csz, 1, invC, out + OFF_CR0 + s, 4, 4, 5, 40);

        if (s < 3) {
            long n2 = csz, sm2 = N2S[s + 1];
            for (int b2 = 0; b2 < 4; ++b2) {
                hipMemcpyAsync(cbufA, coeff(s + 1, b2), (size_t)sm2 * sizeof(float2),
                               hipMemcpyDeviceToDevice, stream);
                fft2d(cbufA, m, -1, stream);
                k_zero<<<gridFor(2 * n2), 256, 0, stream>>>((float*)cbufC, 2 * n2);
                k_expand_scatter<<<gridFor(sm2), 256, 0, stream>>>(cbufA, m, cbufC);
                fft2d(cbufC, n, +1, stream);
                k_parent_cols<<<gridFor(n2), 256, 0, stream>>>(cbufC, rrAr[b2], riAr[b2],
                                                               mgAr[b2], n2);
                k_init_acc<<<1, 1, 0, stream>>>(acc1);
                k_reduce<<<gridFor(n2), 256, 0, stream>>>(mgAr[b2], 1, n2, SCP(SC_ZERO), acc1);
                k_sscal<<<1, 1, 0, stream>>>(SCP(SC_PMEAN + b2), acc1, 0, invC);
            }
            GramSet Yp = mkSet();   // parents: mag - mean
            GramSet Yrp = mkSet();  // rparents: [rr0..3, ri0..3]
            for (int b2 = 0; b2 < 4; ++b2) {
                Yp.p[b2] = mgAr[b2]; Yp.sub[b2] = SCP(SC_PMEAN + b2); Yp.stride[b2] = 1;
                Yrp.p[b2] = rrAr[b2]; Yrp.stride[b2] = 1;
                Yrp.p[4 + b2] = riAr[b2]; Yrp.stride[4 + b2] = 1;
            }
            runGram(Xc, Yp, csz, 0, invC, out + OFF_CX0 + s, 4, 4, 4, 16);
            runGram(Xrc, Yrp, csz, 0, invC, out + OFF_CRX0 + s, 4, 8, 4, 32);
        } else {
            // rparents = upsampled low residual + 4 rolls  (n == 128)
            k_r2c_sub<<<gridFor(4096), 256, 0, stream>>>(loPlane, 1, SCP(SC_ZERO), cbufA, 4096);
            fft2d(cbufA, 64, -1, stream);
            k_zero<<<gridFor(2 * csz), 256, 0, stream>>>((float*)cbufC, 2 * csz);
            k_expand_scatter<<<gridFor(4096), 256, 0, stream>>>(cbufA, 64, cbufC);
            fft2d(cbufC, 128, +1, stream);
            k_c2r<<<gridFor(csz), 256, 0, stream>>>(mgAr[0], cbufC, csz);
            k_make_rolled<<<gridFor(csz), 256, 0, stream>>>(mgAr[0], 128, rolled);
            GramSet Y5 = mkSet();
            for (int q = 0; q < 5; ++q) { Y5.p[q] = rolled + (long)q * csz; Y5.stride[q] = 1; }
            runGram(Xrc, Y5, csz, 0, invC, out + OFF_CRX0 + s, 4, 5, 4, 32);
            runGram(Y5, Y5, csz, 1, 4.0f * invC, out + OFF_CR0 + 4, 5, 5, 5, 40);
        }
    }
}